// SpatialAttentionModule_57071525430028
// MI455X (gfx1250) — compile-verified
//
#include <hip/hip_runtime.h>
#include <math.h>

// Spatial attention: channel mean/max pool -> 7x7 conv -> sigmoid -> x*M
// B=16 C=256 H=W=128. Memory-bound: ~512-768 MiB DRAM traffic dominates.

typedef float f4  __attribute__((ext_vector_type(4)));
typedef float v2f __attribute__((ext_vector_type(2)));
typedef float v8f __attribute__((ext_vector_type(8)));

#define B_  16
#define C_  256
#define H_  128
#define W_  128
#define HW_ (H_ * W_)

// ---------------- Kernel 1: channel mean + max pool ----------------
// 65536 threads, each reduces 4 consecutive (h,w) positions over 256 channels.
// Coalesced: per channel a wave touches 512 contiguous bytes. RT loads so x
// lands in the 192MB L2 for reuse by the multiply kernel.
__global__ void pool_kernel(const float* __restrict__ x,
                            float* __restrict__ avg,
                            float* __restrict__ mxp) {
    int gid = blockIdx.x * blockDim.x + threadIdx.x;   // 0..65535
    int b   = gid >> 12;                               // 4096 f4-groups / image
    int hw  = (gid & 4095) << 2;
    const float* p = x + (size_t)b * C_ * HW_ + hw;

    f4 s = {0.0f, 0.0f, 0.0f, 0.0f};
    f4 m = {-INFINITY, -INFINITY, -INFINITY, -INFINITY};
    #pragma unroll 4
    for (int c = 0; c < C_; ++c) {
        f4 v = *(const f4*)(p + (size_t)c * HW_);
        s += v;
        m.x = fmaxf(m.x, v.x);
        m.y = fmaxf(m.y, v.y);
        m.z = fmaxf(m.z, v.z);
        m.w = fmaxf(m.w, v.w);
    }
    s *= (1.0f / 256.0f);
    int o = b * HW_ + hw;
    *(f4*)(avg + o) = s;
    *(f4*)(mxp + o) = m;
}

// ---------------- Kernel 2: 7x7 conv (2->1 ch) + sigmoid via WMMA ----------
// Each wave computes 16 consecutive output pixels of one row.
// D = A x B + C with K accumulated in 25 chunks of 4 over the 98-tap stencil
// (2 ch * 7 * 7, padded to 100).  A[16x4]: weight chunk, identical rows.
// B[4x16]: column n = im2col data for pixel n.  ISA VGPR layouts (f32):
//   A: lanes 0-15 -> {V0:K0, V1:K1}, lanes 16-31 -> {V0:K2, V1:K3}
//   B: lane%16 = N, same K split across lane halves
// => every lane uses taps t0=4k+2*half, t1=t0+1; result for pixel n sits in
// D[0] of lane n (replicated in both halves since all A rows equal).
__device__ __forceinline__ float tapval(int t, int b, int y, int px,
                                        const float* __restrict__ avg,
                                        const float* __restrict__ mxp) {
    int tc = (t < 98) ? t : 0;          // clamp pad taps to a safe address
    int i  = tc / 49;                   // 0 = avg, 1 = max
    int r  = tc - 49 * i;
    int ky = r / 7;
    int kx = r - 7 * ky;
    int yy = y  + ky - 3;
    int xx = px + kx - 3;
    bool ok = (t < 98) && (yy >= 0) && (yy < H_) && (xx >= 0) && (xx < W_);
    int yyc = yy < 0 ? 0 : (yy > H_ - 1 ? H_ - 1 : yy);
    int xxc = xx < 0 ? 0 : (xx > W_ - 1 ? W_ - 1 : xx);
    const float* p = i ? mxp : avg;
    float v = p[b * HW_ + yyc * W_ + xxc];
    return ok ? v : 0.0f;               // branchless: EXEC stays all-1s
}

__global__ void conv_sigmoid_kernel(const float* __restrict__ wgt,
                                    const float* __restrict__ avg,
                                    const float* __restrict__ mxp,
                                    float* __restrict__ Mout) {
    __shared__ float wsm[100];
    int tid = threadIdx.x;
    if (tid < 100) wsm[tid] = (tid < 98) ? wgt[tid] : 0.0f;
    __syncthreads();

    int lane = tid & 31;
    int wave = blockIdx.x * (blockDim.x >> 5) + (tid >> 5);
    int pix0 = wave << 4;               // 16 pixels per wave
    int b    = pix0 >> 14;              // HW_ = 2^14
    int rem  = pix0 & (HW_ - 1);
    int y    = rem >> 7;                // W_ = 2^7
    int x0   = rem & (W_ - 1);

    int half = lane >> 4;
    int n    = lane & 15;
    int px   = x0 + n;

    v8f acc = {0.0f, 0.0f, 0.0f, 0.0f, 0.0f, 0.0f, 0.0f, 0.0f};
    for (int k4 = 0; k4 < 25; ++k4) {
        int t0 = 4 * k4 + 2 * half;
        int t1 = t0 + 1;
        v2f a, bm;
        a.x  = wsm[t0];
        a.y  = wsm[t1];
        bm.x = tapval(t0, b, y, px, avg, mxp);
        bm.y = tapval(t1, b, y, px, avg, mxp);
        acc = __builtin_amdgcn_wmma_f32_16x16x4_f32(
            /*neg_a=*/false, a, /*neg_b=*/false, bm,
            /*c_mod=*/(short)0, acc, /*reuse_a=*/false, /*reuse_b=*/false);
    }
    float z = acc[0];                   // out[pixel n] (replicated over rows)
    // sigmoid via v_exp_f32 + single v_rcp_f32 (avoids the IEEE div expansion)
    float m = __builtin_amdgcn_rcpf(1.0f + __expf(-z));
    if (lane < 16) Mout[b * HW_ + y * W_ + px] = m;
}

// ---------------- Kernel 3: out = x * broadcast(M) ----------------
// NT load for x (last use; still hits L2 if resident from kernel 1),
// NT store for out (pure streaming, keep L2 for x / M).
__global__ void mul_kernel(const float* __restrict__ x,
                           const float* __restrict__ Mm,
                           float* __restrict__ out) {
    size_t gid = (size_t)blockIdx.x * blockDim.x + threadIdx.x; // 0..2^24-1
    size_t f   = gid << 2;                 // flat float index, f4-aligned
    int b  = (int)(f >> 22);               // C_*HW_ = 2^22
    int hw = (int)(f & (size_t)(HW_ - 1));
    f4 xv = __builtin_nontemporal_load((const f4*)(x + f));
    f4 mv = *(const f4*)(Mm + (size_t)b * HW_ + hw);   // reused x256 -> RT
    f4 ov = xv * mv;
    __builtin_nontemporal_store(ov, (f4*)(out + f));
}

extern "C" void kernel_launch(void* const* d_in, const int* in_sizes, int n_in,
                              void* d_out, int out_size, void* d_ws, size_t ws_size,
                              hipStream_t stream) {
    const float* x = (const float*)d_in[0];   // [16,256,128,128]
    const float* w = (const float*)d_in[1];   // [1,2,7,7] = 98 floats
    float* out = (float*)d_out;

    float* wsf = (float*)d_ws;                // 3 MiB used
    float* avg = wsf;
    float* mxp = wsf + (size_t)B_ * HW_;
    float* Mm  = wsf + 2 * (size_t)B_ * HW_;

    // 1) pool: 65536 threads (B*HW/4)
    pool_kernel<<<256, 256, 0, stream>>>(x, avg, mxp);
    // 2) conv+sigmoid: 16384 waves * 16 px = 262144 outputs
    conv_sigmoid_kernel<<<2048, 256, 0, stream>>>(w, avg, mxp, Mm);
    // 3) multiply: 2^24 f4 groups, one per thread
    mul_kernel<<<65536, 256, 0, stream>>>(x, Mm, out);
}